// GNO2d_6219112645352
// MI455X (gfx1250) — compile-verified
//
#include <hip/hip_runtime.h>
#include <math.h>

typedef __attribute__((ext_vector_type(2))) float v2f;
typedef __attribute__((ext_vector_type(8))) float v8f;

#define TWO_PI 6.28318530717958647692f

__device__ __forceinline__ v8f wmma4(v2f a, v2f b, v8f c) {
  // D(16x16) += A(16x4) * B(4x16), f32, wave32
  return __builtin_amdgcn_wmma_f32_16x16x4_f32(
      false, a, false, b, (short)0, c, false, false);
}

// ---------------------------------------------------------------------------
// Kernel 1: forward mode projection.
// One block per (b,ci). Produces Xm[(b*32+ci)][slot(32)][mx(16)][z(2)][re/im].
// slot 0..15  <-> t-mode 0..15      (for w1)
// slot 16..31 <-> t-mode 112..127   (for w2), i.e. mode = slot-32 (mod 128)
// ---------------------------------------------------------------------------
__global__ __launch_bounds__(192)
void fno_fwd_modes(const float* __restrict__ x, float* __restrict__ Xm) {
  const int bc   = blockIdx.x;           // b*32 + ci
  const int tid  = threadIdx.x;
  const int wave = tid >> 5;             // 0..5
  const int lane = tid & 31;
  const int lo   = lane & 15;
  const int hi   = lane >> 4;

  __shared__ float sS[3][16][257];       // P-DFT planes for current 16-row t-chunk
  __shared__ float sEx[256][32];         // X twiddles e^{-i2pi*mx*x/256}: [x][2mx]=cos,[2mx+1]=-sin
  __shared__ float sFc[3][16][33];       // per-chunk X-projection result
  __shared__ float sXp[4][128][16];      // [(z*2+c)][t][mx] combined complex Xp

  for (int idx = tid; idx < 256 * 32; idx += 192) {
    int xx = idx >> 5, n = idx & 31, mx = n >> 1;
    float th = (float)((mx * xx) & 255) * (TWO_PI / 256.0f);
    float s, c;
    __sincosf(th, &s, &c);
    sEx[xx][n] = (n & 1) ? -s : c;
  }
  __syncthreads();

  const long base  = (long)bc * (128L * 256L * 3L);
  const int plane = wave >> 1;           // 0..2
  const int ntile = wave & 1;            // 0..1

  for (int chunk = 0; chunk < 8; ++chunk) {
    const int t0 = chunk * 16;
    // Load x and do the 3-point P-DFT (z=0 real; z=1 complex)
    for (int idx = tid; idx < 16 * 256; idx += 192) {
      int tl = idx >> 8, xx = idx & 255;
      const float* p = x + base + ((long)(t0 + tl) * 256 + xx) * 3;
      float p0 = p[0], p1 = p[1], p2 = p[2];
      sS[0][tl][xx] = p0 + p1 + p2;                 // z=0 (real)
      sS[1][tl][xx] = p0 - 0.5f * (p1 + p2);        // z=1 re
      sS[2][tl][xx] = 0.86602540378f * (p2 - p1);   // z=1 im
    }
    __syncthreads();

    // X-projection GEMM: sFc[plane][m][n] = sum_x sS[plane][m][x] * sEx[x][n]
    {
      v8f d = {};
      for (int k0 = 0; k0 < 256; k0 += 4) {
        int k = k0 + 2 * hi;
        v2f a = { sS[plane][lo][k], sS[plane][lo][k + 1] };
        v2f b = { sEx[k][ntile * 16 + lo], sEx[k + 1][ntile * 16 + lo] };
        d = wmma4(a, b, d);
      }
      for (int r = 0; r < 8; ++r)
        sFc[plane][r + 8 * hi][ntile * 16 + lo] = d[r];
    }
    __syncthreads();

    // Combine the three real planes into complex Xp for these 16 t rows
    for (int idx = tid; idx < 16 * 16; idx += 192) {
      int tl = idx >> 4, mx = idx & 15;
      int t = t0 + tl;
      sXp[0][t][mx] = sFc[0][tl][2 * mx];                              // z0 re
      sXp[1][t][mx] = sFc[0][tl][2 * mx + 1];                          // z0 im
      sXp[2][t][mx] = sFc[1][tl][2 * mx] - sFc[2][tl][2 * mx + 1];     // z1 re
      sXp[3][t][mx] = sFc[1][tl][2 * mx + 1] + sFc[2][tl][2 * mx];     // z1 im
    }
    __syncthreads();
  }

  // T-projection: modes[slot][mx] = sum_t Xp[t][mx] * e^{-i*2pi*mode*t/128}
  // 8 jobs: (z, mtile, re/im)
  for (int job = wave; job < 8; job += 6) {
    const int z = job >> 2, mtile = (job >> 1) & 1, cout = job & 1;
    const int slot = mtile * 16 + lo;
    const int mode = (slot < 16) ? slot : slot - 32;
    v8f d = {};
    for (int k0 = 0; k0 < 128; k0 += 4) {
      int t = k0 + 2 * hi;
      float s0, c0, s1, c1;
      __sincosf((float)((mode * t + 4096) & 127) * (TWO_PI / 128.0f), &s0, &c0);
      __sincosf((float)((mode * (t + 1) + 4096) & 127) * (TWO_PI / 128.0f), &s1, &c1);
      v2f bre = { sXp[2 * z + 0][t][lo], sXp[2 * z + 0][t + 1][lo] };
      v2f bim = { sXp[2 * z + 1][t][lo], sXp[2 * z + 1][t + 1][lo] };
      v2f ac  = { c0, c1 };
      v2f as  = { s0, s1 };
      v2f asn = { -s0, -s1 };
      if (cout == 0) {            // re = sum XpRe*cos + XpIm*sin
        d = wmma4(ac, bre, d);
        d = wmma4(as, bim, d);
      } else {                    // im = sum XpIm*cos - XpRe*sin
        d = wmma4(ac, bim, d);
        d = wmma4(asn, bre, d);
      }
    }
    for (int r = 0; r < 8; ++r) {
      int slotM = mtile * 16 + r + 8 * hi;
      long o = ((((long)bc * 32 + slotM) * 16 + lo) * 2 + z) * 2 + cout;
      Xm[o] = d[r];
    }
  }
}

// ---------------------------------------------------------------------------
// Kernel 2: per-mode complex channel mixing (tiny: 67 MFLOP total, VALU).
// One block per mode point (mt,mx,z); thread = (b,co).
// ---------------------------------------------------------------------------
__global__ __launch_bounds__(256)
void fno_mix(const float* __restrict__ Xm,
             const float* __restrict__ w1re, const float* __restrict__ w1im,
             const float* __restrict__ w2re, const float* __restrict__ w2im,
             float* __restrict__ Ym) {
  const int blk = blockIdx.x;            // 1024 = 32*16*2
  const int mt = blk >> 5;
  const int mx = (blk >> 1) & 15;
  const int z  = blk & 1;
  const int b  = threadIdx.x >> 5;
  const int co = threadIdx.x & 31;

  const float* wre;
  const float* wim;
  int mtw;
  if (mt < 16) { wre = w1re; wim = w1im; mtw = mt; }
  else         { wre = w2re; wim = w2im; mtw = mt - 16; }

  float ar = 0.0f, ai = 0.0f;
  for (int ci = 0; ci < 32; ++ci) {
    long xo = ((((long)(b * 32 + ci) * 32 + mt) * 16 + mx) * 2 + z) * 2;
    float xr = Xm[xo], xi = Xm[xo + 1];
    long wo = (((long)(ci * 32 + co) * 16 + mtw) * 16 + mx) * 3 + z;
    float wr = wre[wo], wi = wim[wo];
    ar += xr * wr - xi * wi;
    ai += xr * wi + xi * wr;
  }
  long yo = ((((long)(b * 32 + co) * 32 + mt) * 16 + mx) * 2 + z) * 2;
  Ym[yo] = ar;
  Ym[yo + 1] = ai;
}

// ---------------------------------------------------------------------------
// Kernel 3: inverse transform. One block per (b,co).
// T-expansion (K=32) -> LDS g[t][mx][z], then X-expansion (K=16) fused with
// the 3-point inverse P transform and the final store.
// ---------------------------------------------------------------------------
__global__ __launch_bounds__(192)
void fno_inv(const float* __restrict__ Ym, float* __restrict__ out) {
  const int bc   = blockIdx.x;           // b*32 + co
  const int tid  = threadIdx.x;
  const int wave = tid >> 5;
  const int lane = tid & 31;
  const int lo   = lane & 15;
  const int hi   = lane >> 4;

  __shared__ float sY[32 * 16 * 2 * 2];  // [slot][mx][z][c]
  __shared__ float sG[4][128][17];       // [(z*2+c)][t][mx], padded for A reads

  for (int idx = tid; idx < 4096; idx += 192)
    sY[idx] = Ym[(long)bc * 4096 + idx];
  __syncthreads();

  // T-expansion: g[t][mx] = sum_slot Y[slot][mx] * e^{+i*2pi*mode*t/128}
  for (int job = wave; job < 32; job += 6) {
    const int tt = job >> 2, z = (job >> 1) & 1, c = job & 1;
    const int t = tt * 16 + lo;          // A row
    v8f d = {};
    for (int k0 = 0; k0 < 32; k0 += 4) {
      int k = k0 + 2 * hi;
      int m0 = (k < 16) ? k : k - 32;
      int m1 = (k + 1 < 16) ? (k + 1) : (k + 1 - 32);
      float s0, c0, s1, c1;
      __sincosf((float)((m0 * t + 4096) & 127) * (TWO_PI / 128.0f), &s0, &c0);
      __sincosf((float)((m1 * t + 4096) & 127) * (TWO_PI / 128.0f), &s1, &c1);
      v2f bre = { sY[((k * 16 + lo) * 2 + z) * 2 + 0],
                  sY[(((k + 1) * 16 + lo) * 2 + z) * 2 + 0] };
      v2f bim = { sY[((k * 16 + lo) * 2 + z) * 2 + 1],
                  sY[(((k + 1) * 16 + lo) * 2 + z) * 2 + 1] };
      v2f ac  = { c0, c1 };
      v2f as  = { s0, s1 };
      v2f asn = { -s0, -s1 };
      if (c == 0) {                      // re = Yre*cos - Yim*sin
        d = wmma4(ac, bre, d);
        d = wmma4(asn, bim, d);
      } else {                           // im = Yre*sin + Yim*cos
        d = wmma4(as, bre, d);
        d = wmma4(ac, bim, d);
      }
    }
    for (int r = 0; r < 8; ++r)
      sG[z * 2 + c][tt * 16 + r + 8 * hi][lo] = d[r];
  }
  __syncthreads();

  // X-expansion + inverse 3-point P transform + store
  const float SC = 1.0f / (128.0f * 256.0f * 3.0f);
  const float SQ3 = 1.73205080757f;
  for (int job = wave; job < 128; job += 6) {
    const int tt = job >> 4, xc = job & 15;
    const int xg = xc * 16 + lo;         // B column
    const int trow = tt * 16 + lo;       // A row
    v8f d0 = {}, d1 = {}, d2 = {};       // h0.re, h1.re, h1.im
    for (int k0 = 0; k0 < 16; k0 += 4) {
      int k = k0 + 2 * hi;
      float s0, c0, s1, c1;
      __sincosf((float)((k * xg) & 255) * (TWO_PI / 256.0f), &s0, &c0);
      __sincosf((float)(((k + 1) * xg) & 255) * (TWO_PI / 256.0f), &s1, &c1);
      v2f a0r = { sG[0][trow][k], sG[0][trow][k + 1] };
      v2f a0i = { sG[1][trow][k], sG[1][trow][k + 1] };
      v2f a1r = { sG[2][trow][k], sG[2][trow][k + 1] };
      v2f a1i = { sG[3][trow][k], sG[3][trow][k + 1] };
      v2f bco = { c0, c1 };
      v2f bsi = { s0, s1 };
      v2f bsn = { -s0, -s1 };
      d0 = wmma4(a0r, bco, d0); d0 = wmma4(a0i, bsn, d0);
      d1 = wmma4(a1r, bco, d1); d1 = wmma4(a1i, bsn, d1);
      d2 = wmma4(a1r, bsi, d2); d2 = wmma4(a1i, bco, d2);
    }
    for (int r = 0; r < 8; ++r) {
      int t = tt * 16 + r + 8 * hi;
      int xx = xc * 16 + lo;
      long o = (((long)bc * 128 + t) * 256 + xx) * 3;
      float h0 = d0[r], h1r = d1[r], h1i = d2[r];
      out[o + 0] = SC * (h0 + 2.0f * h1r);
      out[o + 1] = SC * (h0 - h1r - SQ3 * h1i);
      out[o + 2] = SC * (h0 - h1r + SQ3 * h1i);
    }
  }
}

// ---------------------------------------------------------------------------
extern "C" void kernel_launch(void* const* d_in, const int* in_sizes, int n_in,
                              void* d_out, int out_size, void* d_ws, size_t ws_size,
                              hipStream_t stream) {
  const float* x   = (const float*)d_in[0];
  const float* w1r = (const float*)d_in[1];
  const float* w1i = (const float*)d_in[2];
  const float* w2r = (const float*)d_in[3];
  const float* w2i = (const float*)d_in[4];
  float* out = (float*)d_out;

  // workspace: forward modes (2 MB) + mixed modes (2 MB)
  float* Xm = (float*)d_ws;              // 256*32*16*2*2 = 524288 floats
  float* Ym = Xm + 524288;               // 524288 floats

  fno_fwd_modes<<<256, 192, 0, stream>>>(x, Xm);
  fno_mix<<<1024, 256, 0, stream>>>(Xm, w1r, w1i, w2r, w2i, Ym);
  fno_inv<<<256, 192, 0, stream>>>(Ym, out);
}